// Beta_TCVAE_71159018160768
// MI455X (gfx1250) — compile-verified
//
#include <hip/hip_runtime.h>
#include <hip/hip_bf16.h>
#include <math.h>

typedef __attribute__((ext_vector_type(2))) float v2f;
typedef __attribute__((ext_vector_type(4))) float v4f;
typedef __attribute__((ext_vector_type(8))) float v8f;

#define kN   1024
#define kD   64
#define kPix 12288
#define kTol 1e-7f
#define kBeta 6.0f
#define kLog2Pi 1.83787706640934548356f
#define kDataset 737280.0f

// ---- workspace layout (float offsets) ----
#define OFF_S     ((size_t)0)                       // N*N   = 1048576
#define OFF_H     (OFF_S   + (size_t)kN * kN)       // N*D
#define OFF_C     (OFF_H   + (size_t)kN * kD)       // N*D
#define OFF_INV   (OFF_C   + (size_t)kN * kD)       // N*D
#define OFF_CSUM  (OFF_INV + (size_t)kN * kD)       // N
#define OFF_MROW  (OFF_CSUM + (size_t)kN)           // N
#define OFF_RSUM  (OFF_MROW + (size_t)kN)           // N
#define OFF_SUMMD (OFF_RSUM + (size_t)kN)           // N
#define OFF_E2    (OFF_SUMMD + (size_t)kN)          // N
#define OFF_PX    (OFF_E2   + (size_t)kN)           // 1024 partials

// K1: per-row precompute of h = 0.5*m^2, c = -0.5*lv - 0.5*log2pi,
//     inv = 1/(exp(lv)+tol), Csum[j] = sum_d c[j,d]
__global__ __launch_bounds__(64) void k1_pre(const float* __restrict__ zmean,
                                             const float* __restrict__ zlogvar,
                                             float* __restrict__ h,
                                             float* __restrict__ c,
                                             float* __restrict__ inv,
                                             float* __restrict__ csum) {
    __shared__ float sh[64];
    const int row = blockIdx.x;
    const int d = threadIdx.x;
    const size_t idx = (size_t)row * kD + d;
    const float zm = zmean[idx];
    const float lv = zlogvar[idx];
    h[idx] = 0.5f * zm * zm;
    const float cv = -0.5f * lv - 0.5f * kLog2Pi;
    c[idx] = cv;
    inv[idx] = 1.0f / (expf(lv) + kTol);
    sh[d] = cv; __syncthreads();
    for (int s = 32; s > 0; s >>= 1) { if (d < s) sh[d] += sh[d + s]; __syncthreads(); }
    if (d == 0) csum[row] = sh[0];
}

// K2: streaming Bernoulli log-likelihood partials. This is the only HBM-bound
// stage (~100MB once => ~4.3us at 23.3 TB/s). b128 non-temporal loads keep
// bytes/instruction high and avoid polluting the L2-resident c/inv/S working
// set; __logf -> single v_log_f32 TRANS op co-executing with the FMAs.
__global__ __launch_bounds__(256) void k2_logpx(const float* __restrict__ target,
                                                const float* __restrict__ xmean,
                                                float* __restrict__ pxpart) {
    __shared__ float sh[256];
    const v4f* __restrict__ t4 = (const v4f*)target;
    const v4f* __restrict__ x4 = (const v4f*)xmean;
    const size_t total4 = (size_t)kN * kPix / 4;   // 3,145,728 vec4
    float acc = 0.0f;
    for (size_t idx = (size_t)blockIdx.x * 256 + threadIdx.x; idx < total4;
         idx += (size_t)gridDim.x * 256) {
        const v4f t = __builtin_nontemporal_load(t4 + idx);
        const v4f x = __builtin_nontemporal_load(x4 + idx);
#pragma unroll
        for (int q = 0; q < 4; ++q) {
            acc += t[q] * __logf(x[q] + kTol)
                 + (1.0f - t[q]) * __logf(1.0f - x[q] + kTol);
        }
    }
    sh[threadIdx.x] = acc; __syncthreads();
    for (int s = 128; s > 0; s >>= 1) {
        if (threadIdx.x < s) sh[threadIdx.x] += sh[threadIdx.x + s];
        __syncthreads();
    }
    if (threadIdx.x == 0) pxpart[blockIdx.x] = sh[0];
}

// K3: S[i,j] = Csum[j] - dot(h[i,:], inv[j,:]) via V_WMMA_F32_16X16X4_F32.
// One wave per 16x16 tile; 8 waves/block; 64x64 tiles total.
// A (16x4 f32, 2 VGPR): lanes 0-15 -> M=0..15 K={kb,kb+1}; lanes 16-31 -> K={kb+2,kb+3}.
// B (4x16 f32, 2 VGPR): same striping with N along lanes (B[k,n] = inv[j0+n, k]).
__global__ __launch_bounds__(256) void k3_gemm_wmma(const float* __restrict__ h,
                                                    const float* __restrict__ inv,
                                                    const float* __restrict__ csum,
                                                    float* __restrict__ S) {
    const int lane = threadIdx.x & 31;
    const int wave = threadIdx.x >> 5;
    const int tile = blockIdx.x * 8 + wave;   // 4096 tiles
    const int ti = tile >> 6;
    const int tj = tile & 63;
    const int i0 = ti * 16;
    const int j0 = tj * 16;
    const int half = lane >> 4;               // 0 or 1
    const int r = lane & 15;
    const int koff = half * 2;

    const float* __restrict__ ha = h   + (size_t)(i0 + r) * kD + koff;
    const float* __restrict__ ba = inv + (size_t)(j0 + r) * kD + koff;

    v8f acc = {};
#pragma unroll
    for (int kb = 0; kb < kD; kb += 4) {
        v2f a, b;
        a.x = ha[kb]; a.y = ha[kb + 1];
        b.x = ba[kb]; b.y = ba[kb + 1];
        acc = __builtin_amdgcn_wmma_f32_16x16x4_f32(
            /*neg_a=*/false, a, /*neg_b=*/false, b,
            /*c_mod=*/(short)0, acc, /*reuse_a=*/false, /*reuse_b=*/false);
    }
    // C/D layout: VGPR v -> row (v + 8*half), col = lane%16
    const float cj = csum[j0 + r];
#pragma unroll
    for (int v = 0; v < 8; ++v) {
        const int row = i0 + v + half * 8;
        S[(size_t)row * kN + (j0 + r)] = cj - acc[v];
    }
}

// K4: per row i of S: m_row[i] = max_j S[i,j]; rsum[i] = sum_j exp(S[i,j]-m_row[i]).
// 256 threads x one b128 load covers the whole 1024-float row; it stays in
// registers across both the max pass and the exp pass (S read exactly once).
__global__ __launch_bounds__(256) void k4_rowstats(const float* __restrict__ S,
                                                   float* __restrict__ mrow,
                                                   float* __restrict__ rsum) {
    __shared__ float sh[256];
    const int i = blockIdx.x;
    const v4f* __restrict__ row4 = (const v4f*)(S + (size_t)i * kN);
    const v4f v = row4[threadIdx.x];
    float m = fmaxf(fmaxf(v[0], v[1]), fmaxf(v[2], v[3]));
    sh[threadIdx.x] = m; __syncthreads();
    for (int s = 128; s > 0; s >>= 1) {
        if (threadIdx.x < s) sh[threadIdx.x] = fmaxf(sh[threadIdx.x], sh[threadIdx.x + s]);
        __syncthreads();
    }
    m = sh[0]; __syncthreads();
    const float acc = __expf(v[0] - m) + __expf(v[1] - m)
                    + __expf(v[2] - m) + __expf(v[3] - m);
    sh[threadIdx.x] = acc; __syncthreads();
    for (int s = 128; s > 0; s >>= 1) {
        if (threadIdx.x < s) sh[threadIdx.x] += sh[threadIdx.x + s];
        __syncthreads();
    }
    if (threadIdx.x == 0) { mrow[i] = m; rsum[i] = sh[0]; }
}

// K5: per (i,d): M[i,d] = max_j (c[j,d] - h[i,d]*inv[j,d]);
//     E[i,d] = sum_j exp(val - M[i,d]).
// Outputs sumMd[i] = sum_d M[i,d] and e2[i] = sum_d E[i,d].
// c/inv (512KB total) stay L2-resident across all 1024 blocks.
__global__ __launch_bounds__(64) void k5_prod(const float* __restrict__ h,
                                              const float* __restrict__ c,
                                              const float* __restrict__ inv,
                                              float* __restrict__ sumMd,
                                              float* __restrict__ e2) {
    __shared__ float sh[64];
    const int i = blockIdx.x;
    const int d = threadIdx.x;
    const float hv = h[(size_t)i * kD + d];
    float m = -INFINITY;
    for (int j = 0; j < kN; ++j) {
        const float val = fmaf(-hv, inv[(size_t)j * kD + d], c[(size_t)j * kD + d]);
        m = fmaxf(m, val);
    }
    float e = 0.0f;
    for (int j = 0; j < kN; ++j) {
        const float val = fmaf(-hv, inv[(size_t)j * kD + d], c[(size_t)j * kD + d]);
        e += __expf(val - m);
    }
    sh[d] = m; __syncthreads();
    for (int s = 32; s > 0; s >>= 1) { if (d < s) sh[d] += sh[d + s]; __syncthreads(); }
    const float sm = sh[0]; __syncthreads();
    sh[d] = e; __syncthreads();
    for (int s = 32; s > 0; s >>= 1) { if (d < s) sh[d] += sh[d + s]; __syncthreads(); }
    if (d == 0) { sumMd[i] = sm; e2[i] = sh[0]; }
}

// K6: final combine. log_pz cancels algebraically:
// loss = -( log_px + (1-beta) * mean_i(log_qz[i] - log_qz_prod[i]) )
__global__ __launch_bounds__(256) void k6_final(const float* __restrict__ pxpart,
                                                const float* __restrict__ mrow,
                                                const float* __restrict__ rsum,
                                                const float* __restrict__ sumMd,
                                                const float* __restrict__ e2,
                                                float* __restrict__ out) {
    __shared__ float sh[256];
    auto reduce = [&](const float* __restrict__ p, int n) -> float {
        float a = 0.0f;
        for (int j = threadIdx.x; j < n; j += 256) a += p[j];
        sh[threadIdx.x] = a; __syncthreads();
        for (int s = 128; s > 0; s >>= 1) {
            if (threadIdx.x < s) sh[threadIdx.x] += sh[threadIdx.x + s];
            __syncthreads();
        }
        const float r = sh[0]; __syncthreads();
        return r;
    };
    const float sum_px = reduce(pxpart, 1024);
    const float sum_m  = reduce(mrow, kN);
    const float s_tot  = reduce(rsum, kN);    // global exp-sum for log_qz
    const float sum_Md = reduce(sumMd, kN);
    const float s2     = reduce(e2, kN);      // global exp-sum for log_qz_prod
    if (threadIdx.x == 0) {
        const float log_px = sum_px / (float)kN;
        const float avg_m  = sum_m  / (float)kN;
        const float avg_Md = sum_Md / (float)kN;
        const float log_nm = logf((float)kN * kDataset);
        // mean_i(log_qz - log_qz_prod)
        const float diff = logf(s_tot) + (float)(kD - 1) * log_nm + avg_m
                         - (float)kD * logf(s2) - avg_Md;
        out[0] = -(log_px + (1.0f - kBeta) * diff);
    }
}

extern "C" void kernel_launch(void* const* d_in, const int* in_sizes, int n_in,
                              void* d_out, int out_size, void* d_ws, size_t ws_size,
                              hipStream_t stream) {
    (void)in_sizes; (void)n_in; (void)out_size; (void)ws_size;
    const float* target = (const float*)d_in[0];
    const float* xmean  = (const float*)d_in[1];
    // d_in[2] = x_log_var: unused by the reference computation
    const float* zmean  = (const float*)d_in[3];
    const float* zlv    = (const float*)d_in[4];

    float* ws    = (float*)d_ws;
    float* S     = ws + OFF_S;
    float* h     = ws + OFF_H;
    float* c     = ws + OFF_C;
    float* inv   = ws + OFF_INV;
    float* csum  = ws + OFF_CSUM;
    float* mrow  = ws + OFF_MROW;
    float* rsum  = ws + OFF_RSUM;
    float* sumMd = ws + OFF_SUMMD;
    float* e2    = ws + OFF_E2;
    float* pxp   = ws + OFF_PX;

    k1_pre<<<kN, kD, 0, stream>>>(zmean, zlv, h, c, inv, csum);
    k2_logpx<<<1024, 256, 0, stream>>>(target, xmean, pxp);
    k3_gemm_wmma<<<512, 256, 0, stream>>>(h, inv, csum, S);
    k4_rowstats<<<kN, 256, 0, stream>>>(S, mrow, rsum);
    k5_prod<<<kN, kD, 0, stream>>>(h, c, inv, sumMd, e2);
    k6_final<<<1, 256, 0, stream>>>(pxp, mrow, rsum, sumMd, e2, (float*)d_out);
}